// GCNTopK2_72095321030885
// MI455X (gfx1250) — compile-verified
//
#include <hip/hip_runtime.h>
#include <hip/hip_bf16.h>

// ---------------------------------------------------------------------------
// GNN (GraphConv + BN + GELU + TopK pool) x2 + linear readout for gfx1250.
// GEMMs: V_WMMA_F32_16X16X4_F32 (16x16 tile / wave32), weight panels staged
// in LDS via GLOBAL_LOAD_ASYNC_TO_LDS_B128 (ASYNCcnt) per 64-deep K chunk.
// ---------------------------------------------------------------------------

#define G_   64
#define NPG_ 1024
#define DEG_ 8
#define NN_  (G_ * NPG_)       // 65536 nodes
#define EE_  (NN_ * DEG_)      // 524288 edges
#define INF_ 128
#define HID_ 256
#define K1_  512
#define K2_  256

typedef __attribute__((ext_vector_type(2))) float v2f;
typedef __attribute__((ext_vector_type(8))) float v8f;

__device__ __forceinline__ v8f wmma_f32_k4(v2f a, v2f b, v8f c) {
  // D = A(16x4) * B(4x16) + C(16x16), fp32, wave32
  return __builtin_amdgcn_wmma_f32_16x16x4_f32(
      /*neg_a=*/false, a, /*neg_b=*/false, b,
      /*c_mod=*/(short)0, c, /*reuse_a=*/false, /*reuse_b=*/false);
}

// Async DMA: 16 bytes global -> LDS, tracked by ASYNCcnt (gfx1250).
// LDS byte offset = low 32 bits of the flat address (aperture mapping).
__device__ __forceinline__ void async_load_b128(const float* gsrc, float* lds_dst) {
  unsigned loff = (unsigned)(unsigned long long)(uintptr_t)lds_dst;
  unsigned long long ga = (unsigned long long)(uintptr_t)gsrc;
  asm volatile("global_load_async_to_lds_b128 %0, %1, off"
               :: "v"(loff), "v"(ga) : "memory");
}
__device__ __forceinline__ void wait_async0() {
  asm volatile("s_wait_asynccnt 0x0" ::: "memory");
}

__device__ __forceinline__ float gelu_exact(float v) {
  return 0.5f * v * (1.0f + erff(v * 0.70710678118654752f));
}

// ---------------------------------------------------------------------------
// Dual-source WMMA GEMM: Out[M,256] = A1[M,K]@W1[K,256] (+ A2[M,K]@W2[K,256])
//                                     + bias, optional GELU + column stats.
// grid = (M/16, 2), block = 256 (8 waves); wave w owns cols [gy*128+w*16,+16).
// Weight panel W[k0:k0+64, gy*128 : +128] staged in 32KB LDS per chunk.
// ---------------------------------------------------------------------------
__global__ void __launch_bounds__(256) mm_dual_wmma(
    const float* __restrict__ A1, const float* __restrict__ W1,
    const float* __restrict__ A2, const float* __restrict__ W2,
    const float* __restrict__ bias, float* __restrict__ Out,
    int M, int K, int do_gelu,
    float* __restrict__ colsum, float* __restrict__ colsq) {
  __shared__ float sW[64 * 128];                 // 32 KB K-chunk of weights
  const int lane = threadIdx.x & 31;
  const int wave = threadIdx.x >> 5;
  const int r0 = blockIdx.x << 4;
  const int c0b = blockIdx.y << 7;               // block's first column
  const int mi = lane & 15;                      // row (A) / col (B) in tile
  const int kk = (lane >> 4) << 1;               // 0 (lanes 0-15) / 2 (16-31)
  const int wcol = (wave << 4) + mi;             // column within 128-wide panel

  v8f acc = {};

  for (int s = 0; s < 2; ++s) {
    const float* A = (s == 0) ? A1 : A2;
    const float* W = (s == 0) ? W1 : W2;
    if (A == nullptr) break;                     // A2 may be null
    const long arow = (long)(r0 + mi) * K;

    for (int k0 = 0; k0 < K; k0 += 64) {
      // ---- async-stage 64x128 weight chunk: 2048 x b128, 8 per thread ----
#pragma unroll
      for (int i = 0; i < 8; ++i) {
        const int q = (i << 8) + (int)threadIdx.x;  // chunk id 0..2047
        const int row = q >> 5;                      // 0..63
        const int c4 = (q & 31) << 2;                // 0..124
        async_load_b128(&W[(long)(k0 + row) * 256 + c0b + c4], &sW[row * 128 + c4]);
      }
      wait_async0();
      __syncthreads();

      // ---- 16 WMMAs on the chunk; A streamed from global, B from LDS ----
#pragma unroll
      for (int k = 0; k < 64; k += 4) {
        __builtin_prefetch(&A[arow + k0 + k + 64], 0, 3);  // global_prefetch_b8
        v2f a, b;
        a.x = A[arow + k0 + k + kk];
        a.y = A[arow + k0 + k + kk + 1];
        b.x = sW[(k + kk) * 128 + wcol];
        b.y = sW[(k + kk + 1) * 128 + wcol];
        acc = wmma_f32_k4(a, b, acc);
      }
      __syncthreads();
    }
  }

  // C/D layout: VGPR r -> M = r (lanes 0-15) or 8+r (lanes 16-31), N = lane&15
  const int rbase = r0 + ((lane >> 4) << 3);
  const int col = c0b + wcol - (wave << 4) + (wave << 4);  // = c0b + wcol
  const float bb = bias[c0b + wcol];
#pragma unroll
  for (int r = 0; r < 8; ++r) {
    float v = acc[r] + bb;
    if (do_gelu) v = gelu_exact(v);
    Out[(long)(rbase + r) * 256 + c0b + wcol] = v;
    if (colsum != nullptr) {
      atomicAdd(&colsum[c0b + wcol], v);
      atomicAdd(&colsq[c0b + wcol], v * v);
    }
  }
  (void)col;
  (void)M;
}

// ---------------------------------------------------------------------------
// Edge scatter: agg[dst] += x[src] (skip masked edges). grid=E, block=F.
// ---------------------------------------------------------------------------
__global__ void edge_agg(const float* __restrict__ x, const int* __restrict__ src,
                         const int* __restrict__ dst, const int* __restrict__ msk,
                         float* __restrict__ agg, int F) {
  const int e = blockIdx.x;
  if (msk != nullptr && msk[e] == 0) return;
  const int s = src[e];
  const int d = dst[e];
  atomicAdd(&agg[(long)d * F + threadIdx.x], x[(long)s * F + threadIdx.x]);
}

// ---------------------------------------------------------------------------
// BN stats finalize + ||p||. 1 block, 256 threads.
// ---------------------------------------------------------------------------
__global__ void stats_finalize(const float* __restrict__ colsum,
                               const float* __restrict__ colsq,
                               const float* __restrict__ g, const float* __restrict__ bt,
                               const float* __restrict__ p,
                               float* __restrict__ scale, float* __restrict__ shift,
                               float* __restrict__ pnorm, float invM) {
  __shared__ float red[256];
  const int t = threadIdx.x;
  const float pv = p[t];
  red[t] = pv * pv;
  __syncthreads();
  for (int s = 128; s > 0; s >>= 1) {
    if (t < s) red[t] += red[t + s];
    __syncthreads();
  }
  if (t == 0) pnorm[0] = sqrtf(red[0]);
  const float mean = colsum[t] * invM;
  const float var = colsq[t] * invM - mean * mean;
  const float sc = g[t] * rsqrtf(var + 1e-5f);
  scale[t] = sc;
  shift[t] = bt[t] - mean * sc;
}

// out = (gelu?)(in*scale + shift); grid = rows, block = 256 (one col each)
__global__ void affine_act(const float* __restrict__ in, float* __restrict__ out,
                           const float* __restrict__ scale,
                           const float* __restrict__ shift, int do_gelu) {
  const long i = (long)blockIdx.x * 256 + threadIdx.x;
  float v = in[i] * scale[threadIdx.x] + shift[threadIdx.x];
  if (do_gelu) v = gelu_exact(v);
  out[i] = v;
}

// score[row] = tanh(dot(h[row,:], p) / ||p||); grid = rows, block = 256
__global__ void score_dot(const float* __restrict__ h, const float* __restrict__ p,
                          const float* __restrict__ pnorm, float* __restrict__ score) {
  __shared__ float red[256];
  const int t = threadIdx.x;
  red[t] = h[(long)blockIdx.x * 256 + t] * p[t];
  __syncthreads();
  for (int s = 128; s > 0; s >>= 1) {
    if (t < s) red[t] += red[t + s];
    __syncthreads();
  }
  if (t == 0) score[blockIdx.x] = tanhf(red[0] / pnorm[0]);
}

// ---------------------------------------------------------------------------
// Per-graph top-k via LDS bitonic sort (descending). grid = G, block = n.
// ---------------------------------------------------------------------------
__global__ void topk_sort(const float* __restrict__ score, int n, int k,
                          int* __restrict__ perm_old, float* __restrict__ vals,
                          int* __restrict__ mapping) {
  __shared__ float sv[1024];
  __shared__ int si[1024];
  const int g = blockIdx.x;
  const int t = threadIdx.x;
  sv[t] = score[g * n + t];
  si[t] = t;
  __syncthreads();
  for (int size = 2; size <= n; size <<= 1) {
    for (int stride = size >> 1; stride > 0; stride >>= 1) {
      const int partner = t ^ stride;
      if (partner > t) {
        const bool desc = ((t & size) == 0);
        const float v0 = sv[t], v1 = sv[partner];
        if (desc ? (v0 < v1) : (v0 > v1)) {
          const int i0 = si[t];
          sv[t] = v1; sv[partner] = v0;
          si[t] = si[partner]; si[partner] = i0;
        }
      }
      __syncthreads();
    }
  }
  if (mapping != nullptr) mapping[g * n + si[t]] = (t < k) ? (g * k + t) : -1;
  if (t < k) {
    perm_old[g * k + t] = g * n + si[t];
    vals[g * k + t] = sv[t];
  }
}

// out[new] = h[old]*val (+ column stats for next BN). grid = Mnew, block = 256
__global__ void gather_pool(const float* __restrict__ h, const int* __restrict__ perm_old,
                            const float* __restrict__ vals, float* __restrict__ out,
                            float* __restrict__ colsum, float* __restrict__ colsq) {
  const int i = blockIdx.x;
  const int t = threadIdx.x;
  const float v = h[(long)perm_old[i] * 256 + t] * vals[i];
  out[(long)i * 256 + t] = v;
  if (colsum != nullptr) {
    atomicAdd(&colsum[t], v);
    atomicAdd(&colsq[t], v * v);
  }
}

__global__ void remap_edges(const int* __restrict__ src, const int* __restrict__ dst,
                            const int* __restrict__ mapping, int* __restrict__ s1,
                            int* __restrict__ d1, int* __restrict__ m1, int E) {
  const int e = blockIdx.x * 256 + threadIdx.x;
  if (e >= E) return;
  const int ns = mapping[src[e]];
  const int nd = mapping[dst[e]];
  const int ok = (ns >= 0) && (nd >= 0);
  s1[e] = ok ? ns : 0;
  d1[e] = ok ? nd : 0;
  m1[e] = ok;
}

// Per-graph readout: xout[g] = [max over k rows | mean over k rows]; grid=G
__global__ void readout(const float* __restrict__ hp, float* __restrict__ xout, int k) {
  const int g = blockIdx.x;
  const int t = threadIdx.x;
  const float* base = hp + (long)g * k * 256 + t;
  float mx = -3.402823466e38f, sm = 0.0f;
  for (int r = 0; r < k; ++r) {
    const float v = base[(long)r * 256];
    mx = fmaxf(mx, v);
    sm += v;
  }
  xout[g * 512 + t] = mx;
  xout[g * 512 + 256 + t] = sm / (float)k;
}

// ---------------------------------------------------------------------------
extern "C" void kernel_launch(void* const* d_in, const int* in_sizes, int n_in,
                              void* d_out, int out_size, void* d_ws, size_t ws_size,
                              hipStream_t stream) {
  const float* x       = (const float*)d_in[0];
  const int*   src     = (const int*)d_in[1];
  const int*   dst     = (const int*)d_in[2];
  const float* W_rel1  = (const float*)d_in[3];
  const float* b_rel1  = (const float*)d_in[4];
  const float* W_root1 = (const float*)d_in[5];
  const float* g1      = (const float*)d_in[6];
  const float* bt1     = (const float*)d_in[7];
  const float* p1      = (const float*)d_in[8];
  const float* g2      = (const float*)d_in[9];
  const float* bt2     = (const float*)d_in[10];
  const float* W_rel2  = (const float*)d_in[11];
  const float* b_rel2  = (const float*)d_in[12];
  const float* W_root2 = (const float*)d_in[13];
  const float* p2      = (const float*)d_in[14];
  const float* Wl      = (const float*)d_in[15];
  const float* bl      = (const float*)d_in[16];
  float* out = (float*)d_out;

  // ---- workspace carve-up (256B aligned) ----
  char* base = (char*)d_ws;
  auto alloc = [&](size_t bytes) -> void* {
    void* p = (void*)base;
    base += (bytes + 255) & ~(size_t)255;
    return p;
  };
  float* agg1    = (float*)alloc((size_t)NN_ * INF_ * 4);
  float* h1      = (float*)alloc((size_t)NN_ * HID_ * 4);
  float* colsum1 = (float*)alloc(256 * 4);
  float* colsq1  = (float*)alloc(256 * 4);
  float* scale1  = (float*)alloc(256 * 4);
  float* shift1  = (float*)alloc(256 * 4);
  float* pnorm1  = (float*)alloc(256);
  float* score1  = (float*)alloc((size_t)NN_ * 4);
  int*   perm1   = (int*)alloc((size_t)G_ * K1_ * 4);
  float* vals1   = (float*)alloc((size_t)G_ * K1_ * 4);
  int*   mapping = (int*)alloc((size_t)NN_ * 4);
  int*   s1      = (int*)alloc((size_t)EE_ * 4);
  int*   d1      = (int*)alloc((size_t)EE_ * 4);
  int*   m1      = (int*)alloc((size_t)EE_ * 4);
  float* hpool1  = (float*)alloc((size_t)G_ * K1_ * HID_ * 4);
  float* colsum2 = (float*)alloc(256 * 4);
  float* colsq2  = (float*)alloc(256 * 4);
  float* scale2  = (float*)alloc(256 * 4);
  float* shift2  = (float*)alloc(256 * 4);
  float* pnorm2  = (float*)alloc(256);
  float* h2in    = (float*)alloc((size_t)G_ * K1_ * HID_ * 4);
  float* agg2    = (float*)alloc((size_t)G_ * K1_ * HID_ * 4);
  float* h2      = (float*)alloc((size_t)G_ * K1_ * HID_ * 4);
  float* score2  = (float*)alloc((size_t)G_ * K1_ * 4);
  int*   perm2   = (int*)alloc((size_t)G_ * K2_ * 4);
  float* vals2   = (float*)alloc((size_t)G_ * K2_ * 4);
  float* hpool2  = (float*)alloc((size_t)G_ * K2_ * HID_ * 4);
  float* x1      = (float*)alloc((size_t)G_ * 512 * 4);
  float* x2      = (float*)alloc((size_t)G_ * 512 * 4);

  const int M1 = NN_;            // 65536
  const int M2 = G_ * K1_;       // 32768

  // ---- zero accumulators (graph-capture-safe async memsets) ----
  hipMemsetAsync(agg1, 0, (size_t)NN_ * INF_ * 4, stream);
  hipMemsetAsync(agg2, 0, (size_t)M2 * HID_ * 4, stream);
  hipMemsetAsync(colsum1, 0, 256 * 4, stream);
  hipMemsetAsync(colsq1, 0, 256 * 4, stream);
  hipMemsetAsync(colsum2, 0, 256 * 4, stream);
  hipMemsetAsync(colsq2, 0, 256 * 4, stream);

  // ---- layer 1: conv -> gelu -> BN -> score -> topk -> pool/readout ----
  edge_agg<<<EE_, INF_, 0, stream>>>(x, src, dst, nullptr, agg1, INF_);
  mm_dual_wmma<<<dim3(M1 / 16, 2), 256, 0, stream>>>(
      agg1, W_rel1, x, W_root1, b_rel1, h1, M1, INF_, /*gelu=*/1, colsum1, colsq1);
  stats_finalize<<<1, 256, 0, stream>>>(colsum1, colsq1, g1, bt1, p1,
                                        scale1, shift1, pnorm1, 1.0f / (float)M1);
  affine_act<<<M1, 256, 0, stream>>>(h1, h1, scale1, shift1, /*gelu=*/0);
  score_dot<<<M1, 256, 0, stream>>>(h1, p1, pnorm1, score1);
  topk_sort<<<G_, NPG_, 0, stream>>>(score1, NPG_, K1_, perm1, vals1, mapping);
  gather_pool<<<M2, 256, 0, stream>>>(h1, perm1, vals1, hpool1, colsum2, colsq2);
  readout<<<G_, 256, 0, stream>>>(hpool1, x1, K1_);
  remap_edges<<<(EE_ + 255) / 256, 256, 0, stream>>>(src, dst, mapping, s1, d1, m1, EE_);

  // ---- layer 2: BN -> gelu -> conv -> score -> topk -> pool/readout ----
  stats_finalize<<<1, 256, 0, stream>>>(colsum2, colsq2, g2, bt2, p2,
                                        scale2, shift2, pnorm2, 1.0f / (float)M2);
  affine_act<<<M2, 256, 0, stream>>>(hpool1, h2in, scale2, shift2, /*gelu=*/1);
  edge_agg<<<EE_, HID_, 0, stream>>>(h2in, s1, d1, m1, agg2, HID_);
  mm_dual_wmma<<<dim3(M2 / 16, 2), 256, 0, stream>>>(
      agg2, W_rel2, h2in, W_root2, b_rel2, h2, M2, HID_, /*gelu=*/0, nullptr, nullptr);
  score_dot<<<M2, 256, 0, stream>>>(h2, p2, pnorm2, score2);
  topk_sort<<<G_, K1_, 0, stream>>>(score2, K1_, K2_, perm2, vals2, nullptr);
  gather_pool<<<G_ * K2_, 256, 0, stream>>>(h2, perm2, vals2, hpool2, nullptr, nullptr);
  readout<<<G_, 256, 0, stream>>>(hpool2, x2, K2_);

  // ---- final: out = (x1 + x2) @ Wl + bl  ==  x1@Wl + x2@Wl + bl ----
  mm_dual_wmma<<<dim3(G_ / 16, 2), 256, 0, stream>>>(
      x1, Wl, x2, Wl, bl, out, G_, 512, /*gelu=*/0, nullptr, nullptr);
}